// GCNModelCMVAE_63110249447564
// MI455X (gfx1250) — compile-verified
//
#include <hip/hip_runtime.h>

// ---------------------------------------------------------------------------
// Problem constants (from reference)
// ---------------------------------------------------------------------------
#define NN 8192      // num nodes
#define FF 512       // features
#define H1 32        // hidden1
#define H2 16        // hidden2
#define EE 131072    // nnz

// workspace layout (float offsets)
#define WS_XW   0                       // [NN, H1]      = 262144
#define WS_H1   262144                  // [NN, H1]      = 262144
#define WS_T    524288                  // 3 x [NN, H2]  = 393216
#define WS_U    917504                  // 3 x [NN, H2]  = 393216
#define WS_Z    1310720                 // [NN, H2]      = 131072
// total = 1441792 floats (~5.8 MB)

typedef float v2f __attribute__((ext_vector_type(2)));
typedef float v8f __attribute__((ext_vector_type(8)));

// D = A(16x4 f32) * B(4x16 f32) + C(16x16 f32)   — CDNA5 V_WMMA_F32_16X16X4_F32
__device__ __forceinline__ v8f wmma4(v2f a, v2f b, v8f c) {
  return __builtin_amdgcn_wmma_f32_16x16x4_f32(
      /*neg_a=*/false, a, /*neg_b=*/false, b,
      /*c_mod=*/(short)0, c, /*reuse_a=*/false, /*reuse_b=*/false);
}

// ---------------------------------------------------------------------------
// zero-fill scratch (h1 + t + u regions) — needed every call (atomic scatter)
// ---------------------------------------------------------------------------
__global__ void k_zero(float* __restrict__ p) {
  int i = blockIdx.x * blockDim.x + threadIdx.x;
  p[i] = 0.0f;
}

// ---------------------------------------------------------------------------
// XW = X @ W0     [8192,512] x [512,32] -> [8192,32], f32 WMMA 16x16x4
// one wave per 16x16 tile; 512 row-tiles * 2 col-tiles = 1024 waves
// ---------------------------------------------------------------------------
__global__ void k_gemm_xw0(const float* __restrict__ X,
                           const float* __restrict__ W0,
                           float* __restrict__ XW) {
  int wave = (blockIdx.x * blockDim.x + threadIdx.x) >> 5;
  int lane = threadIdx.x & 31;
  int tileM = wave >> 1;          // 0..511
  int tileN = wave & 1;           // 0..1
  int m    = lane & 15;
  int half = lane >> 4;

  const float* xrow  = X  + (tileM * 16 + m) * FF + 2 * half;   // A: M=m, K base
  const float* wbase = W0 + (2 * half) * H1 + tileN * 16 + m;   // B: N=m, K base

  v8f acc = {};
  for (int k = 0; k < FF; k += 4) {
    v2f a = *(const v2f*)(xrow + k);          // X[row, k+2h], X[row, k+2h+1]
    v2f b;
    b.x = wbase[k * H1];                      // W0[k+2h    , col]
    b.y = wbase[k * H1 + H1];                 // W0[k+2h + 1, col]
    acc = wmma4(a, b, acc);
  }
  // C layout: VGPR v, lanes0-15 -> M=v ; lanes16-31 -> M=v+8 ; N = lane%16
  float* o = XW + (tileM * 16 + 8 * half) * H1 + tileN * 16 + m;
  #pragma unroll
  for (int v = 0; v < 8; ++v) o[v * H1] = acc[v];
}

// ---------------------------------------------------------------------------
// h1 += scatter( val[e] * XW[cols[e], f] ) over rows[e]; one thread per (e,f)
// ---------------------------------------------------------------------------
__global__ void k_spmm_h1(const int* __restrict__ rows,
                          const int* __restrict__ cols,
                          const float* __restrict__ vals,
                          const float* __restrict__ XW,
                          float* __restrict__ h1) {
  int idx = blockIdx.x * blockDim.x + threadIdx.x;   // E*32 threads
  int e = idx >> 5;
  int f = idx & 31;
  float v = vals[e] * XW[cols[e] * H1 + f];
  unsafeAtomicAdd(&h1[rows[e] * H1 + f], v);         // global_atomic_add_f32
}

__global__ void k_relu(float* __restrict__ p) {
  int i = blockIdx.x * blockDim.x + threadIdx.x;
  p[i] = fmaxf(p[i], 0.0f);
}

// ---------------------------------------------------------------------------
// t[h] = h1 @ W(h)   [8192,32] x [32,16] -> [8192,16], f32 WMMA, head=blockIdx.y
// ---------------------------------------------------------------------------
__global__ void k_gemm_heads(const float* __restrict__ h1,
                             const float* __restrict__ W1,
                             const float* __restrict__ W2,
                             const float* __restrict__ W3,
                             float* __restrict__ t) {
  int head = blockIdx.y;
  const float* W = (head == 0) ? W1 : ((head == 1) ? W2 : W3);
  float* out = t + head * (NN * H2);

  int wave = (blockIdx.x * blockDim.x + threadIdx.x) >> 5;  // 0..511 row-tile
  int lane = threadIdx.x & 31;
  int m    = lane & 15;
  int half = lane >> 4;

  const float* arow  = h1 + (wave * 16 + m) * H1 + 2 * half;
  const float* wbase = W + (2 * half) * H2 + m;

  v8f acc = {};
  #pragma unroll
  for (int k = 0; k < H1; k += 4) {
    v2f a = *(const v2f*)(arow + k);
    v2f b;
    b.x = wbase[k * H2];
    b.y = wbase[k * H2 + H2];
    acc = wmma4(a, b, acc);
  }
  float* o = out + (wave * 16 + 8 * half) * H2 + m;
  #pragma unroll
  for (int v = 0; v < 8; ++v) o[v * H2] = acc[v];
}

// ---------------------------------------------------------------------------
// u[h] = A @ t[h]  scatter-atomics; one thread per (e, f), head = blockIdx.y
// ---------------------------------------------------------------------------
__global__ void k_spmm_heads(const int* __restrict__ rows,
                             const int* __restrict__ cols,
                             const float* __restrict__ vals,
                             const float* __restrict__ t,
                             float* __restrict__ u) {
  int head = blockIdx.y;
  int idx = blockIdx.x * blockDim.x + threadIdx.x;   // E*16 threads
  int e = idx >> 4;
  int f = idx & 15;
  float v = vals[e] * t[head * (NN * H2) + cols[e] * H2 + f];
  unsafeAtomicAdd(&u[head * (NN * H2) + rows[e] * H2 + f], v);
}

// ---------------------------------------------------------------------------
// z = u1 + s2 * ( exp(softmax(u2)) + s1 * 0.1*exp(softmax(u3)) )
// 16 lanes per row (H2=16), shuffle reductions within half-wave
// ---------------------------------------------------------------------------
__global__ void k_combine(const float* __restrict__ u,
                          const float* __restrict__ s1,
                          const float* __restrict__ s2,
                          float* __restrict__ z) {
  int idx = blockIdx.x * blockDim.x + threadIdx.x;   // NN*H2 threads
  float u1 = u[idx];
  float a2 = u[NN * H2 + idx];
  float a3 = u[2 * NN * H2 + idx];

  float m2 = a2, m3 = a3;
  #pragma unroll
  for (int off = 8; off; off >>= 1) {
    m2 = fmaxf(m2, __shfl_xor(m2, off, 16));
    m3 = fmaxf(m3, __shfl_xor(m3, off, 16));
  }
  float e2 = __expf(a2 - m2);
  float e3 = __expf(a3 - m3);
  float sum2 = e2, sum3 = e3;
  #pragma unroll
  for (int off = 8; off; off >>= 1) {
    sum2 += __shfl_xor(sum2, off, 16);
    sum3 += __shfl_xor(sum3, off, 16);
  }
  float sm2 = e2 / sum2;                 // softmax(u2)
  float sm3 = e3 / sum3;                 // softmax(u3)
  float z_en  = __expf(sm2);
  float z_he  = 0.1f * __expf(sm3);
  float z_enn = z_en + s1[idx] * z_he;
  z[idx] = u1 + s2[idx] * z_enn;
}

// ---------------------------------------------------------------------------
// out = z @ z.T   [8192,16] -> [8192,8192]; f32 WMMA, K=16 in 4 steps.
// HBM-write bound: 268 MB / 23.3 TB/s ~= 11.5 us floor. z (512 KB) lives in L2.
// Output (268 MB) > L2 (192 MB) and is never re-read -> NON-TEMPORAL stores
// (th:TH_STORE_NT) so the streaming writes don't evict the hot z tiles.
// B[k,n] = z[tj*16+n, k] -> with B layout (N=lane%16, K=2*(lane/16)+g) the
// per-lane B load is the SAME contiguous b64 pattern as A.
// ---------------------------------------------------------------------------
__global__ void k_zzt(const float* __restrict__ z, float* __restrict__ out) {
  int wave = (blockIdx.x * blockDim.x + threadIdx.x) >> 5;  // 0..262143
  int lane = threadIdx.x & 31;
  int ti = wave >> 9;          // 512 col tiles per row of tiles
  int tj = wave & 511;
  int m    = lane & 15;
  int half = lane >> 4;

  const float* arow = z + (ti * 16 + m) * H2 + 2 * half;
  const float* brow = z + (tj * 16 + m) * H2 + 2 * half;

  v8f acc = {};
  #pragma unroll
  for (int k = 0; k < H2; k += 4) {
    v2f a = *(const v2f*)(arow + k);
    v2f b = *(const v2f*)(brow + k);
    acc = wmma4(a, b, acc);
  }

  long col = tj * 16 + m;
  long row0 = ti * 16 + 8 * half;
  #pragma unroll
  for (int v = 0; v < 8; ++v)
    __builtin_nontemporal_store(acc[v], &out[(row0 + v) * (long)NN + col]);
}

// ---------------------------------------------------------------------------
// launch
// ---------------------------------------------------------------------------
extern "C" void kernel_launch(void* const* d_in, const int* in_sizes, int n_in,
                              void* d_out, int out_size, void* d_ws, size_t ws_size,
                              hipStream_t stream) {
  const float* features = (const float*)d_in[0];   // [NN, FF]
  const int*   adj_rows = (const int*)  d_in[1];   // [EE]
  const int*   adj_cols = (const int*)  d_in[2];   // [EE]
  const float* adj_val  = (const float*)d_in[3];   // [EE]
  const float* W0       = (const float*)d_in[4];   // [FF, H1]
  const float* W1       = (const float*)d_in[5];   // [H1, H2]
  const float* W2       = (const float*)d_in[6];
  const float* W3       = (const float*)d_in[7];
  const float* sample_1 = (const float*)d_in[8];   // [NN, H2]
  const float* sample_2 = (const float*)d_in[9];   // [NN, H2]
  float* out = (float*)d_out;                      // [NN*NN]

  float* ws = (float*)d_ws;
  float* XW = ws + WS_XW;
  float* h1 = ws + WS_H1;
  float* t  = ws + WS_T;
  float* u  = ws + WS_U;
  float* z  = ws + WS_Z;

  // 1) zero h1 + t + u (atomic-accumulated regions): 1,048,576 floats
  k_zero<<<4096, 256, 0, stream>>>(ws + WS_H1);

  // 2) XW = X @ W0 : 1024 waves = 128 blocks x 8 waves
  k_gemm_xw0<<<128, 256, 0, stream>>>(features, W0, XW);

  // 3) h1 scatter: E*32 = 4,194,304 threads
  k_spmm_h1<<<16384, 256, 0, stream>>>(adj_rows, adj_cols, adj_val, XW, h1);

  // 4) relu in place: 262,144 elements
  k_relu<<<1024, 256, 0, stream>>>(h1);

  // 5) three head GEMMs: 512 waves each, grid.y = head
  k_gemm_heads<<<dim3(64, 3), 256, 0, stream>>>(h1, W1, W2, W3, t);

  // 6) head SpMMs: E*16 = 2,097,152 threads per head
  k_spmm_heads<<<dim3(8192, 3), 256, 0, stream>>>(adj_rows, adj_cols, adj_val, t, u);

  // 7) softmax + reparameterization combine: 131,072 threads
  k_combine<<<512, 256, 0, stream>>>(u, sample_1, sample_2, z);

  // 8) out = z @ z.T : 262,144 tiles = 32,768 blocks x 8 waves
  k_zzt<<<32768, 256, 0, stream>>>(z, out);
}